// DSGATConv_49572512530557
// MI455X (gfx1250) — compile-verified
//
#include <hip/hip_runtime.h>
#include <math.h>

#define N_NODES 50000
#define N_EDGES 800000
#define IN_NODE 256
#define IN_EDGE 64
#define HEADS 4
#define OUT 32
#define HD (HEADS * OUT)   // 128
#define NEG_SLOPE 0.2f

typedef __attribute__((ext_vector_type(2))) float v2f;
typedef __attribute__((ext_vector_type(8))) float v8f;

// ---------------------------------------------------------------- zero init
__global__ void __launch_bounds__(256) k_zero(float* __restrict__ out,
                                              float* __restrict__ denom) {
    int i = blockIdx.x * 256 + threadIdx.x;
    if (i < N_NODES * HD) out[i] = 0.f;
    if (i < N_NODES * HEADS) denom[i] = 0.f;
}

// ------------------------------------------------- fold attn_e into W_e: v[64][4]
__global__ void k_prep_v(const float* __restrict__ We,
                         const float* __restrict__ attn_e,
                         float* __restrict__ v) {
    int t = threadIdx.x;          // 256 threads: k = t/4, h = t%4
    int k = t >> 2, h = t & 3;
    float s = 0.f;
#pragma unroll
    for (int d = 0; d < OUT; ++d)
        s += We[k * HD + h * OUT + d] * attn_e[h * OUT + d];
    v[k * HEADS + h] = s;
}

// ------------------------------------------- node GEMM: feat = node_feat @ W
// One wave computes a full 16x128 output strip: 8 accumulators, so each
// b64 A-load feeds 8 independent v_wmma_f32_16x16x4_f32 (no D->A/B hazard,
// full pipelining, 8x less A traffic than tile-per-wave).
__global__ void __launch_bounds__(128) k_node_gemm(const float* __restrict__ A,
                                                   const float* __restrict__ W,
                                                   float* __restrict__ feat) {
    int wave = (blockIdx.x * 128 + threadIdx.x) >> 5;  // M tile id
    int lane = threadIdx.x & 31;
    if (wave >= N_NODES / 16) return;                  // 3125 M tiles (exact)
    int g  = lane >> 4;          // K group (0: K=0,1  1: K=2,3)
    int mn = lane & 15;          // M index for A, N index for B

    const float* arow  = A + (wave * 16 + mn) * IN_NODE + 2 * g;  // A[M][k0+2g..+1]
    const float* bbase = W + 2 * g * HD + mn;                     // W[k0+2g][n0]

    v8f acc[8] = {};
#pragma unroll 2
    for (int kk = 0; kk < IN_NODE / 4; ++kk) {
        v2f a;
        a.x = arow[4 * kk];
        a.y = arow[4 * kk + 1];
        const float* b0 = bbase + (4 * kk) * HD;
#pragma unroll
        for (int tn = 0; tn < 8; ++tn) {
            v2f b;
            b.x = b0[tn * 16];        // W[k0+2g  ][tn*16+mn]
            b.y = b0[tn * 16 + HD];   // W[k0+2g+1][tn*16+mn]
            acc[tn] = __builtin_amdgcn_wmma_f32_16x16x4_f32(
                /*neg_a=*/false, a, /*neg_b=*/false, b,
                /*c_mod=*/(short)0, acc[tn], /*reuse_a=*/false, /*reuse_b=*/false);
        }
    }
    // D layout: VGPR r -> row (r + 8*g), col = lane&15 (+ tn*16)
    float* drow = feat + (wave * 16 + 8 * g) * HD + mn;
#pragma unroll
    for (int tn = 0; tn < 8; ++tn) {
#pragma unroll
        for (int r = 0; r < 8; ++r)
            drow[r * HD + tn * 16] = acc[tn][r];
    }
}

// -------------------------------------------- el/er: per-(node,head) dot(32)
__global__ void __launch_bounds__(256) k_elr(const float* __restrict__ feat,
                                             const float* __restrict__ attn_l,
                                             const float* __restrict__ attn_r,
                                             float* __restrict__ el,
                                             float* __restrict__ er) {
    int wid  = (blockIdx.x * 256 + threadIdx.x) >> 5;  // (node,head) pair
    int lane = threadIdx.x & 31;
    if (wid >= N_NODES * HEADS) return;
    int n = wid >> 2, h = wid & 3;
    float f  = feat[n * HD + h * OUT + lane];
    float pl = f * attn_l[h * OUT + lane];
    float pr = f * attn_r[h * OUT + lane];
#pragma unroll
    for (int off = 16; off > 0; off >>= 1) {
        pl += __shfl_xor(pl, off, 32);
        pr += __shfl_xor(pr, off, 32);
    }
    if (lane == 0) { el[wid] = pl; er[wid] = pr; }
}

// ------------------- per-edge logits: ee + el[src] + er[dst] -> leaky -> exp
// one wave per edge; streams edge_feat once (the only big HBM read)
__global__ void __launch_bounds__(256) k_edge(const float* __restrict__ ef,
                                              const float* __restrict__ v,
                                              const float* __restrict__ el,
                                              const float* __restrict__ er,
                                              const int* __restrict__ src,
                                              const int* __restrict__ dst,
                                              float* __restrict__ wbuf,
                                              float* __restrict__ denom) {
    int e    = (blockIdx.x * 256 + threadIdx.x) >> 5;
    int lane = threadIdx.x & 31;
    if (e >= N_EDGES) return;
    float x0 = ef[e * IN_EDGE + lane];
    float x1 = ef[e * IN_EDGE + 32 + lane];
    const float* va = v + lane * HEADS;
    const float* vb = v + (32 + lane) * HEADS;
    float p0 = x0 * va[0] + x1 * vb[0];
    float p1 = x0 * va[1] + x1 * vb[1];
    float p2 = x0 * va[2] + x1 * vb[2];
    float p3 = x0 * va[3] + x1 * vb[3];
#pragma unroll
    for (int off = 16; off > 0; off >>= 1) {
        p0 += __shfl_xor(p0, off, 32);
        p1 += __shfl_xor(p1, off, 32);
        p2 += __shfl_xor(p2, off, 32);
        p3 += __shfl_xor(p3, off, 32);
    }
    if (lane < 4) {
        float ee = (lane == 0) ? p0 : (lane == 1) ? p1 : (lane == 2) ? p2 : p3;
        int s = src[e], d = dst[e];
        float l = ee + el[s * HEADS + lane] + er[d * HEADS + lane];
        l = (l >= 0.f) ? l : NEG_SLOPE * l;
        float w = __expf(l);                 // softmax w/o max-shift: |l| <~ 6
        wbuf[e * HEADS + lane] = w;
        atomicAdd(denom + d * HEADS + lane, w);
    }
}

// ------------------- scatter: out[dst] += feat[src] * (w/denom[dst]) per head
__global__ void __launch_bounds__(256) k_scatter(const float* __restrict__ feat,
                                                 const float* __restrict__ wbuf,
                                                 const float* __restrict__ denom,
                                                 const int* __restrict__ src,
                                                 const int* __restrict__ dst,
                                                 float* __restrict__ out) {
    int e    = (blockIdx.x * 256 + threadIdx.x) >> 5;
    int lane = threadIdx.x & 31;
    if (e >= N_EDGES) return;
    int s = src[e], d = dst[e];
    float a0 = wbuf[e * HEADS + 0] / denom[d * HEADS + 0];
    float a1 = wbuf[e * HEADS + 1] / denom[d * HEADS + 1];
    float a2 = wbuf[e * HEADS + 2] / denom[d * HEADS + 2];
    float a3 = wbuf[e * HEADS + 3] / denom[d * HEADS + 3];
    const float* fs = feat + s * HD;
    float* od = out + d * HD;
    atomicAdd(od + lane,      fs[lane]      * a0);
    atomicAdd(od + 32 + lane, fs[32 + lane] * a1);
    atomicAdd(od + 64 + lane, fs[64 + lane] * a2);
    atomicAdd(od + 96 + lane, fs[96 + lane] * a3);
}

// ---------------------------------------------------------------- final ReLU
__global__ void __launch_bounds__(256) k_relu(float* __restrict__ out) {
    int i = blockIdx.x * 256 + threadIdx.x;
    if (i < N_NODES * HD) {
        float x = out[i];
        out[i] = x > 0.f ? x : 0.f;
    }
}

extern "C" void kernel_launch(void* const* d_in, const int* in_sizes, int n_in,
                              void* d_out, int out_size, void* d_ws, size_t ws_size,
                              hipStream_t stream) {
    const float* node_feat = (const float*)d_in[0];
    const float* edge_feat = (const float*)d_in[1];
    const float* W         = (const float*)d_in[2];
    const float* We        = (const float*)d_in[3];
    const float* attn_l    = (const float*)d_in[4];
    const float* attn_r    = (const float*)d_in[5];
    const float* attn_e    = (const float*)d_in[6];
    const int*   src       = (const int*)d_in[7];
    const int*   dst       = (const int*)d_in[8];
    float* out = (float*)d_out;

    // workspace layout (all 256B-aligned by construction)
    char* p = (char*)d_ws;
    float* feat  = (float*)p; p += (size_t)N_NODES * HD * 4;      // 25.6 MB
    float* el    = (float*)p; p += (size_t)N_NODES * HEADS * 4;   // 0.8 MB
    float* er    = (float*)p; p += (size_t)N_NODES * HEADS * 4;   // 0.8 MB
    float* wbuf  = (float*)p; p += (size_t)N_EDGES * HEADS * 4;   // 12.8 MB
    float* denom = (float*)p; p += (size_t)N_NODES * HEADS * 4;   // 0.8 MB
    float* v     = (float*)p;                                     // 1 KB

    k_zero<<<(N_NODES * HD + 255) / 256, 256, 0, stream>>>(out, denom);
    k_prep_v<<<1, 256, 0, stream>>>(We, attn_e, v);
    k_node_gemm<<<(N_NODES / 16 + 3) / 4, 128, 0, stream>>>(node_feat, W, feat);
    k_elr<<<(N_NODES * HEADS) / 8, 256, 0, stream>>>(feat, attn_l, attn_r, el, er);
    k_edge<<<(N_EDGES + 7) / 8, 256, 0, stream>>>(edge_feat, v, el, er, src, dst, wbuf, denom);
    k_scatter<<<(N_EDGES + 7) / 8, 256, 0, stream>>>(feat, wbuf, denom, src, dst, out);
    k_relu<<<(N_NODES * HD + 255) / 256, 256, 0, stream>>>(out);
}